// Factorization_Machine_82205674045607
// MI455X (gfx1250) — compile-verified
//
#include <hip/hip_runtime.h>
#include <cstdint>

// ---- problem constants (from reference) ------------------------------------
#define N_USERS    6040
#define N_MOVIES   3952
#define NUM_INPUTS (N_USERS + N_MOVIES)   // 9992
#define FACT_NUM   16
#define BATCH_N    4194304
#define NV4        (NUM_INPUTS / 4)       // 2498 float4s = 39968 bytes

// typed, address-space-qualified pointers for the async LDS-copy builtin
typedef int v4i __attribute__((ext_vector_type(4)));
typedef __attribute__((address_space(1))) v4i* g_v4i_ptr;   // global
typedef __attribute__((address_space(3))) v4i* l_v4i_ptr;   // LDS

__global__ __launch_bounds__(256) void fm_fwd_kernel(
    const int*   __restrict__ x,      // (BATCH, 3) int32
    const float* __restrict__ emb,    // (NUM_INPUTS, 16) f32
    const float* __restrict__ w,      // (1, NUM_INPUTS) f32
    const float* __restrict__ bias,   // (1,) f32
    float*       __restrict__ out)    // [out | recommended], 2*BATCH f32
{
    __shared__ float s_w[NUM_INPUTS]; // 39.0 KB staged linear table
    __shared__ float s_ib;            // interaction + bias scalar

    const int tid = threadIdx.x;

    // ---- Stage linear_w into LDS via CDNA5 async global->LDS b128 copies ---
#if __has_builtin(__builtin_amdgcn_global_load_async_to_lds_b128)
    for (int i = tid; i < NV4; i += 256) {
        __builtin_amdgcn_global_load_async_to_lds_b128(
            (g_v4i_ptr)(uintptr_t)(w + 4 * i),
            (l_v4i_ptr)(uint32_t)(uintptr_t)(&s_w[4 * i]),
            /*offset=*/0, /*cpol=*/0);
    }
#if __has_builtin(__builtin_amdgcn_s_wait_asynccnt)
    __builtin_amdgcn_s_wait_asynccnt(0);
#else
    asm volatile("s_wait_asynccnt 0" ::: "memory");
#endif
#else
    // inline-asm fallback: VGLOBAL async load, SADDR form:
    //   global_load_async_to_lds_b128 vdst(lds byte addr), voffset, s[base]
    {
        unsigned long long gbase = (unsigned long long)(uintptr_t)w;
        for (int i = tid; i < NV4; i += 256) {
            unsigned lds_off = (unsigned)(uintptr_t)(&s_w[4 * i]);
            unsigned goff    = 16u * (unsigned)i;
            asm volatile("global_load_async_to_lds_b128 %0, %1, %2"
                         :: "v"(lds_off), "v"(goff), "s"(gbase)
                         : "memory");
        }
        asm volatile("s_wait_asynccnt 0" ::: "memory");
    }
#endif

    // ---- interaction scalar (depends only on emb rows 0,1) -----------------
    if (tid == 0) {
        const float c0 = (float)(NUM_INPUTS - 2);  // 9990
        const float c1 = 2.0f;
        float acc = 0.0f;
#pragma unroll
        for (int f = 0; f < FACT_NUM; ++f) {
            float a  = emb[f];
            float b  = emb[FACT_NUM + f];
            float se = c0 * a + c1 * b;
            float sq = c0 * a * a + c1 * b * b;
            acc += se * se - sq;
        }
        s_ib = 0.5f * acc + bias[0];
    }
    __syncthreads();

    const float ib = s_ib;
    const int4*  __restrict__ xv   = (const int4*)x;
    float4*      __restrict__ outv = (float4*)out;
    float4*      __restrict__ recv = (float4*)(out + BATCH_N);

    const int nquad   = BATCH_N / 4;           // 1,048,576
    const int gstride = gridDim.x * 256;

    for (int q = blockIdx.x * 256 + tid; q < nquad; q += gstride) {
        // 4 rows = 12 ints = three b128 loads (fully coalesced, 16B aligned)
        int4 e0 = xv[3 * q + 0];
        int4 e1 = xv[3 * q + 1];
        int4 e2 = xv[3 * q + 2];

        // field decode: row i -> (u, m, r)
        float l0 = s_w[e0.x] + s_w[N_USERS + e0.y] + ib;   // r0 = e0.z
        float l1 = s_w[e0.w] + s_w[N_USERS + e1.x] + ib;   // r1 = e1.y
        float l2 = s_w[e1.z] + s_w[N_USERS + e1.w] + ib;   // r2 = e2.x
        float l3 = s_w[e2.y] + s_w[N_USERS + e2.z] + ib;   // r3 = e2.w

        float4 o;
        o.x = 1.0f / (1.0f + __expf(-l0));
        o.y = 1.0f / (1.0f + __expf(-l1));
        o.z = 1.0f / (1.0f + __expf(-l2));
        o.w = 1.0f / (1.0f + __expf(-l3));

        float4 rc;
        rc.x = (e0.z >= 3) ? 1.0f : 0.0f;
        rc.y = (e1.y >= 3) ? 1.0f : 0.0f;
        rc.z = (e2.x >= 3) ? 1.0f : 0.0f;
        rc.w = (e2.w >= 3) ? 1.0f : 0.0f;

        outv[q] = o;   // b128 store
        recv[q] = rc;  // b128 store
    }
}

extern "C" void kernel_launch(void* const* d_in, const int* in_sizes, int n_in,
                              void* d_out, int out_size, void* d_ws, size_t ws_size,
                              hipStream_t stream) {
    const int*   x    = (const int*)  d_in[0];   // (BATCH,3) int32
    const float* emb  = (const float*)d_in[1];   // (9992,16) f32
    const float* w    = (const float*)d_in[2];   // (1,9992)  f32
    const float* bias = (const float*)d_in[3];   // (1,)      f32
    float*       out  = (float*)d_out;           // 2*BATCH f32

    (void)in_sizes; (void)n_in; (void)out_size; (void)d_ws; (void)ws_size;

    fm_fwd_kernel<<<1024, 256, 0, stream>>>(x, emb, w, bias, out);
}